// SdkBenchmark25PtStencilModel_3083786518617
// MI455X (gfx1250) — compile-verified
//
#include <hip/hip_runtime.h>
#include <hip/hip_bf16.h>
#include <stdint.h>

// -------------------------------------------------------------------------
// 25-pt (radius-4, 3-axis) wave stencil, 10 leap-frog steps, 192^3 fp32.
// Memory-bound: ~140MB/step compulsory HBM traffic => ~6us @ 23.3TB/s;
// cur/prev/vp (85MB) stay L2-resident (192MB), src streams NT.
// fp32 VALU math; CDNA5 async global->LDS copies (ASYNCcnt) double-buffer
// the halo'd X-Y plane; z-taps ride a 9-deep per-thread register window.
// -------------------------------------------------------------------------

#define NXD 192
#define NTOT (NXD * NXD * NXD)
#define TXD 32
#define TYD 8
#define RAD 4
#define PXD 40              // TXD + 2*RAD
#define PYD 16              // TYD + 2*RAD
#define PLANE (PXD * PYD)   // 640 floats
#define CZ 24
#define NTHR 256

// stencil coefficients (dx = 20 -> dx^2 = 400)
__device__ constexpr float LAPC = (float)(3.0 * (-205.0 / 72.0 / 400.0));
__device__ constexpr float CC1  = (float)( 8.0 / 5.0   / 400.0);
__device__ constexpr float CC2  = (float)(-1.0 / 5.0   / 400.0);
__device__ constexpr float CC3  = (float)( 8.0 / 315.0 / 400.0);
__device__ constexpr float CC4  = (float)(-1.0 / 560.0 / 400.0);

// One async fp32 global->LDS copy: lds byte-offset in VGPR, 32-bit global
// byte-offset in VGPR, 64-bit base in SGPR pair (GVS addressing mode).
__device__ __forceinline__ void async_ld_f32(uint32_t lds_off, const float* base,
                                             uint32_t goff) {
  asm volatile("global_load_async_to_lds_b32 %0, %1, %2"
               :: "v"(lds_off), "v"(goff), "s"(base)
               : "memory");
}

__device__ __forceinline__ void wait_async_le3() {
  asm volatile("s_wait_asynccnt 0x3" ::: "memory");
}

__global__ void __launch_bounds__(NTHR)
stencil_step(const float* __restrict__ vp, float* __restrict__ prevnext,
             const float* __restrict__ cur, const float* __restrict__ src) {
  __shared__ float smem[2 * PLANE + 32];

  const int tx  = threadIdx.x, ty = threadIdx.y;
  const int tid = ty * TXD + tx;
  const int bx0 = blockIdx.x * TXD;
  const int by0 = blockIdx.y * TYD;
  const int z0  = blockIdx.z * CZ;
  const int x   = bx0 + tx;
  const int y   = by0 + ty;

  const uint32_t lds_base = (uint32_t)(uintptr_t)(&smem[0]);
  const uint32_t dump_off = lds_base + (uint32_t)(2 * PLANE + (tid & 31)) * 4u;

  // ---- per-lane async-copy constants (z-invariant; hoisted out of z loop) --
  // Exactly 3 unconditional async-load instructions per wave per plane keeps
  // the per-wave ASYNCcnt schedule fixed: wait<=3 retires exactly plane z.
  uint32_t loff0[3], loff1[3], goffb[3], zstep[3];
#pragma unroll
  for (int k = 0; k < 3; ++k) {
    int e  = tid + k * NTHR;          // 0..767
    int py = e / PXD;
    int px = e - py * PXD;
    int gy = by0 + py - RAD;
    int gx = bx0 + px - RAD;
    bool inr = (e < PLANE) & (gx >= 0) & (gx < NXD) & (gy >= 0) & (gy < NXD);
    uint32_t cell = (uint32_t)(py * PXD + px) * 4u;
    loff0[k] = inr ? (lds_base + cell) : dump_off;                  // slot 0
    loff1[k] = inr ? (lds_base + (uint32_t)PLANE * 4u + cell) : dump_off; // slot 1
    goffb[k] = inr ? 4u * (uint32_t)(gy * NXD + gx) : 0u;
    zstep[k] = inr ? 4u * (uint32_t)(NXD * NXD) : 0u;               // 0 => goff stays 0
  }

  // Zero LDS once: halo cells outside the grid are never overwritten and thus
  // provide the reference's zero boundary condition for x/y taps.
  for (int k = tid; k < 2 * PLANE + 32; k += NTHR) smem[k] = 0.0f;
  __syncthreads();  // DS stores retired before async writes can land

  // Prime the pipeline: plane z0 in flight.
  {
    uint32_t zg = (uint32_t)z0;
#pragma unroll
    for (int k = 0; k < 3; ++k)
      async_ld_f32((z0 & 1) ? loff1[k] : loff0[k], cur, goffb[k] + zg * zstep[k]);
  }

  // Per-thread z-direction register window: cur[z-4..z+4] at (y,x).
  const int colbase = y * NXD + x;
  float w[9];
#pragma unroll
  for (int d = 0; d < 8; ++d) {
    int zg = z0 - RAD + d;
    w[d] = (zg >= 0 && zg < NXD) ? cur[zg * (NXD * NXD) + colbase] : 0.0f;
  }
  w[8] = 0.0f;

  for (int z = z0; z < z0 + CZ; ++z) {
    // Stream next plane into the other LDS slot while we compute this one.
    int zn = z + 1; if (zn > NXD - 1) zn = NXD - 1;   // clamped, never read if OOR
    {
      uint32_t zg = (uint32_t)zn;
#pragma unroll
      for (int k = 0; k < 3; ++k)
        async_ld_f32((zn & 1) ? loff1[k] : loff0[k], cur, goffb[k] + zg * zstep[k]);
    }

    int z4 = z + RAD;
    w[8] = (z4 < NXD) ? cur[z4 * (NXD * NXD) + colbase] : 0.0f;

    wait_async_le3();   // this wave's plane-z copies done (3 newest = plane z+1)
    __syncthreads();    // every wave's plane-z copies done

    const float* pl = &smem[(z & 1) * PLANE];
    const int row = (ty + RAD) * PXD + (tx + RAD);

    float c  = w[4];
    float t1 = pl[row - 1] + pl[row + 1] + pl[row - PXD]     + pl[row + PXD]     + w[3] + w[5];
    float t2 = pl[row - 2] + pl[row + 2] + pl[row - 2 * PXD] + pl[row + 2 * PXD] + w[2] + w[6];
    float t3 = pl[row - 3] + pl[row + 3] + pl[row - 3 * PXD] + pl[row + 3 * PXD] + w[1] + w[7];
    float t4 = pl[row - 4] + pl[row + 4] + pl[row - 4 * PXD] + pl[row + 4 * PXD] + w[0] + w[8];
    float lap = LAPC * c + CC1 * t1 + CC2 * t2 + CC3 * t3 + CC4 * t4;

    int   idx = z * (NXD * NXD) + colbase;
    float s   = __builtin_nontemporal_load(src + idx);  // streamed once: NT, spare L2
    float up  = 2.0f * c - prevnext[idx] + vp[idx] * lap + s;
    prevnext[idx] = up;    // prev and next alias: pointwise read-before-write

#pragma unroll
    for (int d = 0; d < 8; ++d) w[d] = w[d + 1];
    __syncthreads();       // all reads of slot z&1 done before it is re-filled
  }
}

// ---------------- init / reductions (deterministic, no float atomics) -------

__global__ void __launch_bounds__(NTHR)
zero4_kernel(float4* __restrict__ p, int n4) {
  int i = blockIdx.x * NTHR + threadIdx.x;
  if (i < n4) p[i] = make_float4(0.0f, 0.0f, 0.0f, 0.0f);
}

__global__ void __launch_bounds__(NTHR)
finalize_partial(const float* __restrict__ outf, const float* __restrict__ ref,
                 float* __restrict__ out, float* __restrict__ psum,
                 float* __restrict__ pmax) {
  __shared__ float ss[NTHR];
  __shared__ float sx[NTHR];
  int t = threadIdx.x;
  int i = blockIdx.x * NTHR + t;
  float v = outf[i];
  float e = v - ref[i];
  out[1 + i] = v;                     // output field at d_out[1..N]
  ss[t] = e * e;
  sx[t] = fabsf(e);
  __syncthreads();
  for (int o = NTHR / 2; o > 0; o >>= 1) {
    if (t < o) { ss[t] += ss[t + o]; sx[t] = fmaxf(sx[t], sx[t + o]); }
    __syncthreads();
  }
  if (t == 0) { psum[blockIdx.x] = ss[0]; pmax[blockIdx.x] = sx[0]; }
}

__global__ void __launch_bounds__(NTHR)
finalize_reduce(const float* __restrict__ psum, const float* __restrict__ pmax,
                int nb, float* __restrict__ out) {
  __shared__ float ss[NTHR];
  __shared__ float sx[NTHR];
  int t = threadIdx.x;
  float s = 0.0f, m = 0.0f;
  for (int i = t; i < nb; i += NTHR) {   // fixed order -> deterministic
    s += psum[i];
    m = fmaxf(m, pmax[i]);
  }
  ss[t] = s; sx[t] = m;
  __syncthreads();
  for (int o = NTHR / 2; o > 0; o >>= 1) {
    if (t < o) { ss[t] += ss[t + o]; sx[t] = fmaxf(sx[t], sx[t + o]); }
    __syncthreads();
  }
  if (t == 0) {
    out[0]        = ss[0] / (float)NTOT;  // loss = mean squared error
    out[1 + NTOT] = sx[0];                // max abs error
  }
}

// ----------------------------- launcher ------------------------------------

extern "C" void kernel_launch(void* const* d_in, const int* in_sizes, int n_in,
                              void* d_out, int out_size, void* d_ws, size_t ws_size,
                              hipStream_t stream) {
  const float* vp  = (const float*)d_in[0];
  const float* src = (const float*)d_in[1];   // [10][192^3]
  const float* ref = (const float*)d_in[2];
  float* out = (float*)d_out;                 // [loss, field(192^3), maxerr]

  float* A    = (float*)d_ws;                 // ping-pong field buffers
  float* B    = A + NTOT;
  float* psum = B + NTOT;                     // 27648 block partials
  float* pmax = psum + (NTOT / NTHR);

  // u(-1) = u(0) = 0
  {
    int n4 = (2 * NTOT) / 4;
    zero4_kernel<<<(n4 + NTHR - 1) / NTHR, NTHR, 0, stream>>>((float4*)A, n4);
  }

  dim3 blk(TXD, TYD, 1);
  dim3 grd(NXD / TXD, NXD / TYD, NXD / CZ);   // 6 x 24 x 8 = 1152 blocks
  for (int t = 0; t < 10; ++t) {
    // carry (prev,cur) -> (cur,upd); upd overwrites prev (pointwise alias).
    float* pn = (t & 1) ? B : A;   // holds prev on entry, next on exit
    float* cu = (t & 1) ? A : B;   // holds cur
    stencil_step<<<grd, blk, 0, stream>>>(vp, pn, cu, src + (size_t)t * NTOT);
  }
  // After t=9 (odd), the newest field lives in B.
  finalize_partial<<<NTOT / NTHR, NTHR, 0, stream>>>(B, ref, out, psum, pmax);
  finalize_reduce<<<1, NTHR, 0, stream>>>(psum, pmax, NTOT / NTHR, out);
}